// LucasKanade_58626303590893
// MI455X (gfx1250) — compile-verified
//
#include <hip/hip_runtime.h>

// ---------------------------------------------------------------------------
// Lucas-Kanade step on MI455X (gfx1250, wave32).
//
//  kernel 1 (lk_prep): per-pixel Sobel Ix/Iy of img1 + (img2-img1) diff,
//                      computed once into d_ws planes (pitch 1920). This gives
//                      22x reuse vs recomputing Sobel per patch element
//                      (100k points x 225 elements gather ~2M pixels worth of
//                      Sobel from only ~1M unique pixels, all L2-resident).
//
//  kernel 2 (lk_solve): one wave handles FIVE points. Rows 3q..3q+2 of a
//                      16x16x4 f32 WMMA hold [Ix; Iy; err] chunks of point q.
//                      The f32 A-matrix (16x4) layout is lane m: VGPR0=K0,
//                      VGPR1=K1 (lanes 16-31: K2,K3); the B-matrix (4x16) has
//                      the identical lane/VGPR mapping with n as lane index,
//                      so passing the SAME register pair as A and B yields
//                      D = E * E^T -- the Gram matrix whose 3x3 diagonal
//                      blocks are exactly (SumIx^2, SumIxIy, SumIy^2,
//                      SumIx*err, SumIy*err) for each packed point.
//                      57 accumulate-chained v_wmma_f32_16x16x4_f32 cover the
//                      225 patch elements. The hot loop is branch-free:
//                      incremental patch addressing (no div-by-15) and
//                      software-pipelined loads that overlap the WMMA chain.
// ---------------------------------------------------------------------------

typedef float v2f __attribute__((ext_vector_type(2)));
typedef float v8f __attribute__((ext_vector_type(8)));

#define IMG_W 1920
#define IMG_H 1080
#define PATCH 15
#define PELEMS (PATCH * PATCH)  // 225

static __device__ __forceinline__ v8f gram_step(float a0, float a1, v8f acc) {
  v2f ab;
  ab.x = a0;
  ab.y = a1;
  // D = A * A^T + C : same registers feed the A-matrix and B-matrix.
  return __builtin_amdgcn_wmma_f32_16x16x4_f32(
      /*neg_a=*/false, ab, /*neg_b=*/false, ab,
      /*c_mod=*/(short)0, acc, /*reuse_a=*/false, /*reuse_b=*/false);
}

// ---------------------------------------------------------------------------
// Kernel 1: Sobel (un-flipped correlation, VALID, matches jax.lax.conv) + diff.
// ---------------------------------------------------------------------------
__global__ __launch_bounds__(256) void lk_prep(const float* __restrict__ img1,
                                               const float* __restrict__ img2,
                                               float* __restrict__ IxP,
                                               float* __restrict__ IyP,
                                               float* __restrict__ errP) {
  int j = blockIdx.x * 16 + (threadIdx.x & 15);
  int i = blockIdx.y * 16 + (threadIdx.x >> 4);
  if (i >= IMG_H || j >= IMG_W) return;
  int o = i * IMG_W + j;
  errP[o] = img2[o] - img1[o];
  if (i < IMG_H - 2 && j < IMG_W - 2) {
    float w00 = img1[o],             w01 = img1[o + 1],             w02 = img1[o + 2];
    float w10 = img1[o + IMG_W],                                    w12 = img1[o + IMG_W + 2];
    float w20 = img1[o + 2 * IMG_W], w21 = img1[o + 2 * IMG_W + 1], w22 = img1[o + 2 * IMG_W + 2];
    // GX = [[2,0,-2],[4,0,-4],[2,0,-2]],  GY = [[2,4,2],[0,0,0],[-2,-4,-2]]
    IxP[o] = 2.0f * (w00 - w02) + 4.0f * (w10 - w12) + 2.0f * (w20 - w22);
    IyP[o] = 2.0f * w00 + 4.0f * w01 + 2.0f * w02 - 2.0f * w20 - 4.0f * w21 - 2.0f * w22;
  } else {
    IxP[o] = 0.0f;
    IyP[o] = 0.0f;
  }
}

// ---------------------------------------------------------------------------
// Kernel 2: 5 points per wave, Gram matrix via v_wmma_f32_16x16x4_f32.
// Lane r = lane&15: point q = r/3 (q==5 on lane 15/31 is padding whose Gram
// row/col 15 is never read), role r%3 selects the Ix/Iy/err plane; half =
// lane>>4 selects K-slots {0,1} vs {2,3}. Element index e = 4*chunk + 2*half+v.
// ---------------------------------------------------------------------------
__global__ __launch_bounds__(256) void lk_solve(const int* __restrict__ pts,
                                                const float* __restrict__ IxP,
                                                const float* __restrict__ IyP,
                                                const float* __restrict__ errP,
                                                float* __restrict__ out, int N) {
  __shared__ float red[8][256];  // 8 waves/block x full 16x16 f32 accumulator

  const int tid   = threadIdx.x;
  const int lane  = tid & 31;
  const int wib   = tid >> 5;
  const int gwave = (blockIdx.x * 256 + tid) >> 5;

  const int r    = lane & 15;
  const int half = lane >> 4;
  const int q    = r / 3;      // 0..4 valid, 5 => padding lane
  const int role = r - q * 3;  // 0=Ix, 1=Iy, 2=err
  const int n    = gwave * 5 + q;

  // One-time guarded coordinate fetch; invalid lanes use (0,0), which keeps
  // every in-loop load in-bounds. Their garbage Gram entries are never read.
  int x = 0, y = 0;
  if (q < 5 && n < N) {
    x = pts[2 * n + 0];
    y = pts[2 * n + 1];
  }
  const float* plane = (role == 0) ? IxP : (role == 1) ? IyP : errP;
  const int base = y * IMG_W + x;

  // Incremental patch addressing: slot pointers advance 4 elements/chunk,
  // wrapping to the next patch row when pj+4 >= 15 (delta = W-11, pj -= 11).
  const float* p0 = plane + base + 2 * half;      // element e = 2*half
  const float* p1 = p0 + 1;                       // element e+1
  int pj0 = 2 * half;
  int pj1 = 2 * half + 1;

  float c0 = *p0;  // software-pipeline: chunk 0 in flight before the loop
  float c1 = *p1;

  v8f acc = {};
#pragma unroll 1
  for (int c = 0; c < 55; ++c) {
    // Advance both slots to the next chunk and issue its loads first...
    const bool w0 = pj0 >= PATCH - 4;
    const bool w1 = pj1 >= PATCH - 4;
    p0 += w0 ? (IMG_W - (PATCH - 4)) : 4;
    p1 += w1 ? (IMG_W - (PATCH - 4)) : 4;
    pj0 = w0 ? pj0 - (PATCH - 4) : pj0 + 4;
    pj1 = w1 ? pj1 - (PATCH - 4) : pj1 + 4;
    const float n0 = *p0;
    const float n1 = *p1;
    // ...then consume the previous chunk, overlapping load latency with the
    // WMMA accumulate chain.
    acc = gram_step(c0, c1, acc);
    c0 = n0;
    c1 = n1;
  }
  acc = gram_step(c0, c1, acc);  // drain chunk 55 (elements 220..223)

  // Tail chunk 56: only element 224 (pi=14, pj=14) is real; K-slots 1..3 are
  // exact zeros so they cannot pollute the sums.
  const float t224 = plane[base + 14 * IMG_W + 14];
  acc = gram_step((half == 0) ? t224 : 0.0f, 0.0f, acc);

  // Spill accumulator: element (m,nn) lives in lane (nn + 16*(m>=8)), VGPR m&7.
#pragma unroll
  for (int v = 0; v < 8; ++v) red[wib][lane * 8 + v] = acc[v];
  __syncthreads();

  if (lane < 5) {
    const int qq = lane;
    const int np = gwave * 5 + qq;
    if (np < N) {
      const int m0 = 3 * qq;
      auto idx = [](int m, int nn) {
        return ((nn + ((m >= 8) ? 16 : 0)) << 3) + (m & 7);
      };
      const float A  = red[wib][idx(m0,     m0)];      // sum Ix*Ix
      const float B  = red[wib][idx(m0,     m0 + 1)];  // sum Ix*Iy
      const float D  = red[wib][idx(m0 + 1, m0 + 1)];  // sum Iy*Iy
      const float b0 = red[wib][idx(m0,     m0 + 2)];  // sum Ix*err
      const float b1 = red[wib][idx(m0 + 1, m0 + 2)];  // sum Iy*err
      const float det = A * D - B * B;
      out[2 * np + 0] = (D * b0 - B * b1) / det;
      out[2 * np + 1] = (A * b1 - B * b0) / det;
    }
  }
}

// ---------------------------------------------------------------------------
// Launcher
// ---------------------------------------------------------------------------
extern "C" void kernel_launch(void* const* d_in, const int* in_sizes, int n_in,
                              void* d_out, int out_size, void* d_ws, size_t ws_size,
                              hipStream_t stream) {
  const float* img1 = (const float*)d_in[0];
  const float* img2 = (const float*)d_in[1];
  const int*   pts  = (const int*)d_in[2];
  const int N = in_sizes[2] / 2;

  float* IxP  = (float*)d_ws;
  float* IyP  = IxP + (size_t)IMG_W * IMG_H;
  float* errP = IyP + (size_t)IMG_W * IMG_H;

  dim3 g1((IMG_W + 15) / 16, (IMG_H + 15) / 16);
  lk_prep<<<g1, 256, 0, stream>>>(img1, img2, IxP, IyP, errP);

  const int waves  = (N + 4) / 5;      // 5 points per wave32
  const int blocks = (waves + 7) / 8;  // 8 waves per 256-thread block
  lk_solve<<<blocks, 256, 0, stream>>>(pts, IxP, IyP, errP, (float*)d_out, N);
}